// Overlaps_18416819765703
// MI455X (gfx1250) — compile-verified
//
#include <hip/hip_runtime.h>
#include <stdint.h>

// Masked pairwise IoU: out[r, c] = IoU(rois0[r], rois1[c]) if (label,batch) match else 0.
// Store-bandwidth / VALU bound (256 MB out, ~300 KB in). CDNA5 paths used:
//  - GLOBAL_LOAD_ASYNC_TO_LDS_B128 + s_wait_asynccnt to stage the column tile (ASYNCcnt path)
//  - explicit global_store_b128 th:TH_STORE_NT for the write-once 256 MB output stream
//    (output > 192 MB L2 -> avoid write-allocate churn), SADDR form: uniform SGPR row base
//    + one 32-bit lane offset, so store addressing costs zero per-row VALU.
//  - v_rcp_f32 for the IoU divide
//  - wave-uniform row data forced onto the scalar (SMEM/SALU) pipe via readfirstlane

typedef __attribute__((ext_vector_type(4))) float v4f;

#define TN 256   // columns per block
#define TM 32    // rows per block
#define RM 8     // rows per thread
#define THREADS 256

__global__ __launch_bounds__(THREADS)
void masked_iou_kernel(const v4f* __restrict__ rois0,
                       const int* __restrict__ lab0,
                       const int* __restrict__ bat0,
                       const v4f* __restrict__ rois1,
                       const int* __restrict__ lab1,
                       const int* __restrict__ bat1,
                       float* __restrict__ out,
                       int N0, int N1)
{
    __shared__ v4f sh_box[TN];
    __shared__ int sh_key[TN];

    const int tid     = threadIdx.x;
    const int colBase = blockIdx.x * TN;
    const int rowBase = blockIdx.y * TM;

    // ---- stage column tile into LDS with CDNA5 async global->LDS copy ----
    {
        int c  = colBase + tid;
        int cc = c < N1 ? c : (N1 - 1);   // clamp so no lane reads OOB
        // Low 32 bits of a generic LDS pointer == address-space(3) byte address.
        uint32_t lds_box = (uint32_t)(uintptr_t)(&sh_box[tid]);
        uint32_t goff    = (uint32_t)cc * 16u;
        unsigned long long gbase = (unsigned long long)(uintptr_t)rois1;
        asm volatile("global_load_async_to_lds_b128 %0, %1, %2"
                     :: "v"(lds_box), "v"(goff), "s"(gbase)
                     : "memory");
        // labels/batches are tiny: fuse to one match key
        sh_key[tid] = (lab1[cc] << 16) | bat1[cc];
        // wait for THIS wave's async copies; barrier below publishes across waves
        asm volatile("s_wait_asynccnt 0" ::: "memory");
    }
    __syncthreads();

    const int tx = tid & 63;        // 64 column groups of 4 -> 512B/wave store bursts
    // ty is constant across each wave32 (waves are 32 consecutive lanes) ->
    // make that explicit so the row side goes down the scalar pipe.
    const int ty = __builtin_amdgcn_readfirstlane(tid >> 6);
    const int c0 = colBase + tx * 4;

    // hoist this thread's 4 column boxes from LDS into registers
    float cx1[4], cy1[4], cx2[4], cy2[4], ca[4];
    int   ck[4];
#pragma unroll
    for (int j = 0; j < 4; ++j) {
        int cj = tx * 4 + j;
        v4f b = sh_box[cj];
        cx1[j] = b.x; cy1[j] = b.y; cx2[j] = b.z; cy2[j] = b.w;
        ca[j]  = (b.z - b.x) * (b.w - b.y);
        ck[j]  = sh_key[cj];
    }

    const int row0 = rowBase + ty * RM;

    // ---- preload all wave-uniform row data (SMEM clause, SALU key/area math) ----
    float rx1[RM], ry1[RM], rx2[RM], ry2[RM], rarea[RM];
    int   rkey[RM];
#pragma unroll
    for (int i = 0; i < RM; ++i) {
        int r  = row0 + i;
        int rr = r < N0 ? r : (N0 - 1);
        v4f a  = rois0[rr];
        rx1[i] = a.x; ry1[i] = a.y; rx2[i] = a.z; ry2[i] = a.w;
        rarea[i] = (a.z - a.x) * (a.w - a.y);
        rkey[i]  = (lab0[rr] << 16) | bat0[rr];
    }

    const bool     vec_ok = ((N1 & 3) == 0) && (c0 + 3 < N1);
    const uint32_t voff   = (uint32_t)c0 * 4u;   // lane-varying store offset (bytes)

#pragma unroll
    for (int i = 0; i < RM; ++i) {
        int r = row0 + i;
        if (r >= N0) continue;

        // hoist area0 + area1[j] per row (1 add/output, SGPR+VGPR operands)
        float sa[4];
#pragma unroll
        for (int j = 0; j < 4; ++j) sa[j] = rarea[i] + ca[j];

        float res[4];
#pragma unroll
        for (int j = 0; j < 4; ++j) {
            float iw = fminf(rx2[i], cx2[j]) - fmaxf(rx1[i], cx1[j]);
            float ih = fminf(ry2[i], cy2[j]) - fmaxf(ry1[i], cy1[j]);
            iw = fmaxf(iw, 0.0f);
            ih = fmaxf(ih, 0.0f);
            float inter = iw * ih;
            float uni   = sa[j] - inter;
            // union<=0 implies inter==0 -> where(union>0,...) select is redundant,
            // but keep max(uni,1e-12) so 0*rcp(0) never produces NaN.
            float iou = inter * __builtin_amdgcn_rcpf(fmaxf(uni, 1e-12f));
            res[j] = (rkey[i] == ck[j]) ? iou : 0.0f;
        }

        if (vec_ok) {
            v4f v = { res[0], res[1], res[2], res[3] };
            // uniform 64-bit row base in SGPRs; NT hint: write-once stream, skip L2 retention
            unsigned long long rowbase =
                (unsigned long long)(uintptr_t)(out + (size_t)r * N1);
            asm volatile("global_store_b128 %0, %1, %2 th:TH_STORE_NT"
                         :: "v"(voff), "v"(v), "s"(rowbase)
                         : "memory");
        } else {
#pragma unroll
            for (int j = 0; j < 4; ++j)
                if (c0 + j < N1) out[(size_t)r * N1 + c0 + j] = res[j];
        }
    }
}

extern "C" void kernel_launch(void* const* d_in, const int* in_sizes, int n_in,
                              void* d_out, int out_size, void* d_ws, size_t ws_size,
                              hipStream_t stream) {
    const v4f* rois0 = (const v4f*)d_in[0];
    const int* lab0  = (const int*)d_in[1];
    const int* bat0  = (const int*)d_in[2];
    const v4f* rois1 = (const v4f*)d_in[3];
    const int* lab1  = (const int*)d_in[4];
    const int* bat1  = (const int*)d_in[5];
    float* out = (float*)d_out;

    int N0 = in_sizes[0] / 4;
    int N1 = in_sizes[3] / 4;

    dim3 grid((N1 + TN - 1) / TN, (N0 + TM - 1) / TM);
    masked_iou_kernel<<<grid, THREADS, 0, stream>>>(rois0, lab0, bat0,
                                                    rois1, lab1, bat1,
                                                    out, N0, N1);
}